// OneConvSAGE_50448685859137
// MI455X (gfx1250) — compile-verified
//
#include <hip/hip_runtime.h>
#include <hip/hip_bf16.h>

typedef __attribute__((ext_vector_type(2))) float v2f;
typedef __attribute__((ext_vector_type(8))) float v8f;

#define IN_FEAT 64
#define OUT_FEAT 64
#define KDIM 128            // 2*IN_FEAT
#define WLDS_STRIDE 130     // pad to avoid LDS bank conflicts

// ---------------------------------------------------------------------------
// Kernel 1: zero msg accumulator (aliased onto d_out) and degree buffer.
// ---------------------------------------------------------------------------
__global__ void sage_zero_kernel(float* __restrict__ msg, float* __restrict__ deg,
                                 int n_msg, int n_nodes) {
    int i = blockIdx.x * blockDim.x + threadIdx.x;
    if (i < n_msg) msg[i] = 0.0f;
    if (i < n_nodes) deg[i] = 0.0f;
}

// ---------------------------------------------------------------------------
// Kernel 2: edge scatter. 16 threads per edge, each handles 4 features
// (float4 gather from h[src], 4x global_atomic_add_f32 into msg[dst]).
// Lane-group leader bumps the degree counter.
// h / msg are 25.6 MB each -> resident in the 192 MB L2, so the random
// traffic never leaves the chip.
// ---------------------------------------------------------------------------
__global__ void sage_scatter_kernel(const float* __restrict__ h,
                                    const int* __restrict__ src,
                                    const int* __restrict__ dst,
                                    float* __restrict__ msg,
                                    float* __restrict__ deg,
                                    int E) {
    int t = blockIdx.x * blockDim.x + threadIdx.x;
    int e = t >> 4;               // edge index
    int fg = (t & 15) << 2;       // feature group: 4 floats
    if (e >= E) return;
    int s = src[e];
    int d = dst[e];
    float4 v = *reinterpret_cast<const float4*>(h + (size_t)s * IN_FEAT + fg);
    float* mp = msg + (size_t)d * IN_FEAT + fg;
    atomicAdd(mp + 0, v.x);
    atomicAdd(mp + 1, v.y);
    atomicAdd(mp + 2, v.z);
    atomicAdd(mp + 3, v.w);
    if ((t & 15) == 0) atomicAdd(deg + d, 1.0f);
}

// ---------------------------------------------------------------------------
// Kernel 3: fused mean + concat + GEMM + bias + ReLU via V_WMMA_F32_16X16X4_F32.
// Block = 256 threads = 8 waves; each wave owns a 16-row x 64-col output
// stripe. K loop split into two fully-unrolled branch-free halves:
//   ks 0..15  : A from h           (k = 0..63)
//   ks 16..31 : A from msg * inv   (k = 64..127)
// fp32 end-to-end. W staged in padded LDS. out rows alias msg rows of the
// same wave only; all A reads feed the accumulators before any store.
// ---------------------------------------------------------------------------
__global__ void __launch_bounds__(256)
sage_wmma_kernel(const float* __restrict__ h,
                 const float* __restrict__ msg,   // == out buffer (aliased)
                 const float* __restrict__ deg,
                 const float* __restrict__ W,     // [64][128] row-major
                 const float* __restrict__ bias,  // [64]
                 float* __restrict__ out,         // [N][64]
                 int N) {
    __shared__ float Wlds[OUT_FEAT * WLDS_STRIDE];

    // cooperative W preload, padded stride
    for (int i = threadIdx.x; i < OUT_FEAT * KDIM; i += blockDim.x) {
        int n = i >> 7;        // /128
        int k = i & 127;
        Wlds[n * WLDS_STRIDE + k] = W[i];
    }
    __syncthreads();

    const int wave = threadIdx.x >> 5;
    const int lane = threadIdx.x & 31;
    const int half = lane >> 4;     // 0: K0/K1 pair, 1: K2/K3 pair
    const int nloc = lane & 15;
    const int off  = half * 2;

    const int row0 = blockIdx.x * 128 + wave * 16;
    int myrow = row0 + nloc;
    int r = myrow < N ? myrow : (N - 1);   // clamp, keep EXEC all-ones

    float dg  = deg[r];
    float inv = 1.0f / fmaxf(dg, 1.0f);

    // single base pointers -> all loads below use immediate offsets
    const float* aH = h   + (size_t)r * IN_FEAT + off;
    const float* aM = msg + (size_t)r * IN_FEAT + off;
    const float* bB = &Wlds[nloc * WLDS_STRIDE + off];

    v8f acc[4];
#pragma unroll
    for (int nt = 0; nt < 4; ++nt) {
        float bv = bias[nt * 16 + nloc];
#pragma unroll
        for (int i = 0; i < 8; ++i) acc[nt][i] = bv;
    }

    // ---- first half: A = h, k = 0..63 ----
#pragma unroll
    for (int ks = 0; ks < 16; ++ks) {
        const int k0 = ks * 4;
        v2f a = *reinterpret_cast<const v2f*>(aH + k0);
#pragma unroll
        for (int nt = 0; nt < 4; ++nt) {
            v2f bf = *reinterpret_cast<const v2f*>(bB + nt * 16 * WLDS_STRIDE + k0);
            acc[nt] = __builtin_amdgcn_wmma_f32_16x16x4_f32(
                false, a, false, bf, (short)0, acc[nt], false, false);
        }
    }

    // ---- second half: A = msg * inv, k = 64..127 ----
#pragma unroll
    for (int ks = 0; ks < 16; ++ks) {
        const int k0 = ks * 4;
        v2f a = *reinterpret_cast<const v2f*>(aM + k0);
        a.x *= inv;
        a.y *= inv;
#pragma unroll
        for (int nt = 0; nt < 4; ++nt) {
            v2f bf = *reinterpret_cast<const v2f*>(bB + nt * 16 * WLDS_STRIDE + IN_FEAT + k0);
            acc[nt] = __builtin_amdgcn_wmma_f32_16x16x4_f32(
                false, a, false, bf, (short)0, acc[nt], false, false);
        }
    }

    // store with ReLU; D layout: c[i] -> row (i + 8*half), col nloc
#pragma unroll
    for (int nt = 0; nt < 4; ++nt) {
#pragma unroll
        for (int i = 0; i < 8; ++i) {
            int row = row0 + i + half * 8;
            if (row < N)
                out[(size_t)row * OUT_FEAT + nt * 16 + nloc] = fmaxf(acc[nt][i], 0.0f);
        }
    }
}

// ---------------------------------------------------------------------------
extern "C" void kernel_launch(void* const* d_in, const int* in_sizes, int n_in,
                              void* d_out, int out_size, void* d_ws, size_t ws_size,
                              hipStream_t stream) {
    const float* h   = (const float*)d_in[0];
    const int*   src = (const int*)d_in[1];
    const int*   dst = (const int*)d_in[2];
    const float* W   = (const float*)d_in[3];
    const float* b   = (const float*)d_in[4];

    const int N = in_sizes[0] / IN_FEAT;
    const int E = in_sizes[1];

    float* msg = (float*)d_out;        // msg_sum accumulates in the output buffer
    float* deg = (float*)d_ws;         // N floats of scratch

    // 1) zero accumulators
    {
        int total = N * IN_FEAT;
        int blocks = (total + 255) / 256;
        sage_zero_kernel<<<blocks, 256, 0, stream>>>(msg, deg, total, N);
    }
    // 2) scatter-add edges
    {
        long long threads = (long long)E * 16;
        int blocks = (int)((threads + 255) / 256);
        sage_scatter_kernel<<<blocks, 256, 0, stream>>>(h, src, dst, msg, deg, E);
    }
    // 3) fused mean/concat/GEMM/bias/ReLU (WMMA fp32)
    {
        int blocks = (N + 127) / 128;
        sage_wmma_kernel<<<blocks, 256, 0, stream>>>(h, msg, deg, W, b, (float*)d_out, N);
    }
}